// Generator_88991722373617
// MI455X (gfx1250) — compile-verified
//
#include <hip/hip_runtime.h>
#include <math.h>

#define BATCH 8
#define STYLE 512
#define NEG 0.2f
#define SQRT2F 1.41421356237309504880f

// K-chunk: 8 input channels x 9 taps = 72 GEMM-K rows per stage
#define KCH  72
#define ASTR 72     // LDS row stride for A (72 mod 64 = 8 -> K/K+2 rows bank-disjoint)
#define BSTR 136    // LDS row stride for B (136 mod 64 = 8 -> same)

typedef __attribute__((ext_vector_type(2))) float v2f;
typedef __attribute__((ext_vector_type(8))) float v8f;

__device__ __forceinline__ float lrelu(float x) {
    return (x >= 0.f ? x : NEG * x) * SQRT2F;
}

// ---------------------------------------------------------------------------
// Generic EqualLinear: out[b,o] = (in[b,:]·w[o,:])*scale + bias[o]*lr (+act)
// ---------------------------------------------------------------------------
__global__ __launch_bounds__(256) void k_linear(
    const float* __restrict__ in, const float* __restrict__ w,
    const float* __restrict__ bias, float* __restrict__ out,
    int IN, int OUT, float scale, float lr, int act)
{
    int idx = blockIdx.x * blockDim.x + threadIdx.x;
    if (idx >= BATCH * OUT) return;
    int b = idx / OUT, o = idx % OUT;
    const float* xr = in + (long)b * IN;
    const float* wr = w + (long)o * IN;
    float acc = 0.f;
    for (int i = 0; i < IN; ++i) acc += xr[i] * wr[i];
    acc = acc * scale + bias[o] * lr;
    if (act) acc = lrelu(acc);
    out[idx] = acc;
}

// PixelNorm over rows of length 512 (one block per batch row)
__global__ __launch_bounds__(256) void k_pixelnorm(
    const float* __restrict__ z, float* __restrict__ out)
{
    __shared__ float red[256];
    int b = blockIdx.x, t = threadIdx.x;
    float a0 = z[b * STYLE + t];
    float a1 = z[b * STYLE + 256 + t];
    red[t] = a0 * a0 + a1 * a1;
    __syncthreads();
    for (int s = 128; s > 0; s >>= 1) {
        if (t < s) red[t] += red[t + s];
        __syncthreads();
    }
    float r = rsqrtf(red[0] / (float)STYLE + 1e-8f);
    out[b * STYLE + t] = a0 * r;
    out[b * STYLE + 256 + t] = a1 * r;
}

// lat = s + lambda * y
__global__ __launch_bounds__(256) void k_mix(
    const float* __restrict__ s, const float* __restrict__ y,
    const float* __restrict__ lam, float* __restrict__ out)
{
    int idx = blockIdx.x * blockDim.x + threadIdx.x;
    if (idx < BATCH * STYLE) out[idx] = s[idx] + lam[0] * y[idx];
}

// d[b,o] = rsqrt( sum_{i,t} (wscale*W[o,i,t]*s[b,i])^2 + 1e-8 )
__global__ __launch_bounds__(256) void k_demod(
    const float* __restrict__ W, const float* __restrict__ s,
    float* __restrict__ d, int IC, int OC)
{
    int idx = blockIdx.x * blockDim.x + threadIdx.x;
    if (idx >= BATCH * OC) return;
    int b = idx / OC, o = idx % OC;
    float wscale = rsqrtf((float)(IC * 9));
    float sum = 0.f;
    for (int i = 0; i < IC; ++i) {
        float si = s[b * IC + i] * wscale;
        const float* wp = W + ((long)o * IC + i) * 9;
        #pragma unroll
        for (int t = 0; t < 9; ++t) {
            float wv = wp[t] * si;
            sum += wv * wv;
        }
    }
    d[idx] = rsqrtf(sum + 1e-8f);
}

// ---------------------------------------------------------------------------
// Modulated 3x3 conv as implicit GEMM on V_WMMA_F32_16X16X4_F32.
//   Block: 256 threads = 8 waves (2 M x 4 N). Block tile 64 oc x 128 pixels.
//   Wave tile 32x32 via 4 accumulators (2x2 fragment tiling).
//   K = IC*9, staged in chunks of 8 ic (72 rows) through LDS.
// UP=false: same-pad conv; epilogue wscale*d*acc + nw*noise + ab, leaky.
// UP=true : conv_transpose(stride2,pad0) == lhs-dilated conv with flipped
//           taps; output res 2R+1; epilogue only wscale*d*acc.
// ---------------------------------------------------------------------------
template <bool UP>
__global__ __launch_bounds__(256) void k_modconv(
    const float* __restrict__ xin, long xin_bstride,
    const float* __restrict__ s,       // [B,IC]
    const float* __restrict__ W,       // [OC,IC,3,3]
    const float* __restrict__ dmod,    // [B,OC]
    const float* __restrict__ noise,   // [RO,RO] (non-up only)
    const float* __restrict__ nw_p,    // scalar  (non-up only)
    const float* __restrict__ ab,      // [OC]    (non-up only)
    float* __restrict__ out,
    int IC, int OC, int R)
{
    const int RO = UP ? (2 * R + 1) : R;
    const int NP = RO * RO;
    const int b  = blockIdx.z;
    const int m0 = blockIdx.y * 64;    // OC is always a multiple of 64
    const int n0 = blockIdx.x * 128;
    const float wscale = rsqrtf((float)(IC * 9));

    __shared__ float As[KCH * ASTR];
    __shared__ float Bs[KCH * BSTR];

    const int tid  = threadIdx.x;
    const int lane = tid & 31;
    const int wid  = tid >> 5;
    const int wm   = (wid & 1) * 32;   // wave M offset in tile
    const int wn   = (wid >> 1) * 32;  // wave N offset in tile

    v8f acc00 = {}, acc01 = {}, acc10 = {}, acc11 = {};

    const float* xb = xin + (long)b * xin_bstride;

    // fixed staging coordinates per thread (no per-element div/mod)
    const int am = tid & 63;           // A column (out-channel)
    const int aq = tid >> 6;           // 0..3 -> il pair
    const int ao = m0 + am;
    const int bn = tid & 127;          // B column (pixel)
    const int bh = tid >> 7;           // 0/1 -> il 0..3 / 4..7
    const int bp = n0 + bn;
    const int byo = bp / RO, bxo = bp % RO;
    const bool bvalid = bp < NP;

    for (int ic0 = 0; ic0 < IC; ic0 += 8) {
        // ---- stage A = raw weights (taps flipped for transpose conv) ----
        {
            const float* wr = W + ((long)ao * IC + ic0) * 9;
            #pragma unroll
            for (int j = 0; j < 2; ++j) {
                int il = aq * 2 + j;
                #pragma unroll
                for (int tap = 0; tap < 9; ++tap) {
                    int tapw = UP ? (8 - tap) : tap;
                    As[(il * 9 + tap) * ASTR + am] = wr[il * 9 + tapw];
                }
            }
            if (ic0 + 8 < IC)
                __builtin_prefetch(wr + 72, 0, 0);   // global_prefetch_b8
        }
        // ---- stage B = s-modulated, zero-padded input gathers ----
        {
            #pragma unroll
            for (int j = 0; j < 4; ++j) {
                int il = bh * 4 + j;
                float sv = bvalid ? s[b * IC + ic0 + il] : 0.f;
                const float* row = xb + (long)(ic0 + il) * R * R;
                #pragma unroll
                for (int tap = 0; tap < 9; ++tap) {
                    int dy = tap / 3, dx = tap % 3;
                    float v = 0.f;
                    if (bvalid) {
                        if (!UP) {
                            int yi = byo + dy - 1, xi = bxo + dx - 1;
                            if (yi >= 0 && yi < R && xi >= 0 && xi < R)
                                v = sv * row[yi * R + xi];
                        } else {
                            int u = byo + dy - 2, w2 = bxo + dx - 2;
                            if (u >= 0 && w2 >= 0 && u <= 2 * R - 2 &&
                                w2 <= 2 * R - 2 && !(u & 1) && !(w2 & 1))
                                v = sv * row[(u >> 1) * R + (w2 >> 1)];
                        }
                    }
                    Bs[(il * 9 + tap) * BSTR + bn] = v;
                }
            }
        }
        __syncthreads();

        // ---- 72 K-rows of fp32 WMMA, 2x2 fragment tiling ----
        const int off = (lane >> 4) << 1;   // lanes 0-15: K+0/1, 16-31: K+2/3
        const int ml  = wm + (lane & 15);
        const int nl  = wn + (lane & 15);
        #pragma unroll 3
        for (int k = 0; k < KCH; k += 4) {
            v2f a0, a1, b0, b1;
            a0.x = As[(k + off) * ASTR + ml];
            a0.y = As[(k + off + 1) * ASTR + ml];
            a1.x = As[(k + off) * ASTR + ml + 16];
            a1.y = As[(k + off + 1) * ASTR + ml + 16];
            b0.x = Bs[(k + off) * BSTR + nl];
            b0.y = Bs[(k + off + 1) * BSTR + nl];
            b1.x = Bs[(k + off) * BSTR + nl + 16];
            b1.y = Bs[(k + off + 1) * BSTR + nl + 16];
            acc00 = __builtin_amdgcn_wmma_f32_16x16x4_f32(
                false, a0, false, b0, (short)0, acc00, false, false);
            acc01 = __builtin_amdgcn_wmma_f32_16x16x4_f32(
                false, a0, false, b1, (short)0, acc01, false, false);
            acc10 = __builtin_amdgcn_wmma_f32_16x16x4_f32(
                false, a1, false, b0, (short)0, acc10, false, false);
            acc11 = __builtin_amdgcn_wmma_f32_16x16x4_f32(
                false, a1, false, b1, (short)0, acc11, false, false);
        }
        __syncthreads();
    }

    // ---- epilogue ----
    float nwv = 0.f;
    if (!UP) nwv = nw_p[0];
    const int hl = lane >> 4;
    #pragma unroll
    for (int tm = 0; tm < 2; ++tm) {
        #pragma unroll
        for (int tn = 0; tn < 2; ++tn) {
            const v8f& ac = tm == 0 ? (tn == 0 ? acc00 : acc01)
                                    : (tn == 0 ? acc10 : acc11);
            int p = n0 + wn + tn * 16 + (lane & 15);
            if (p >= NP) continue;
            #pragma unroll
            for (int vi = 0; vi < 8; ++vi) {
                int o = m0 + wm + tm * 16 + vi + 8 * hl;
                float val = ac[vi] * wscale * dmod[b * OC + o];
                if (!UP) {
                    val += nwv * noise[p] + ab[o];
                    val = lrelu(val);
                }
                out[((long)b * OC + o) * NP + p] = val;
            }
        }
    }
}

// ---------------------------------------------------------------------------
// Blur(K2D*4, pad 1) on transpose-conv output (RT=2R+1 -> R2=2R),
// fused with noise injection + bias + leaky.
// ---------------------------------------------------------------------------
__global__ __launch_bounds__(256) void k_blur_act(
    const float* __restrict__ T, int OC, int RT, int R2,
    const float* __restrict__ noise, const float* __restrict__ nw_p,
    const float* __restrict__ ab, float* __restrict__ out)
{
    long idx = (long)blockIdx.x * blockDim.x + threadIdx.x;
    long total = (long)BATCH * OC * R2 * R2;
    if (idx >= total) return;
    int x = idx % R2;
    int y = (idx / R2) % R2;
    int o = (idx / ((long)R2 * R2)) % OC;
    int b = idx / ((long)R2 * R2 * OC);
    const float k1[4] = {1.f, 3.f, 3.f, 1.f};
    const float* Tb = T + ((long)b * OC + o) * RT * RT;
    float sum = 0.f;
    #pragma unroll
    for (int ky = 0; ky < 4; ++ky) {
        int yi = y + ky - 1;
        if (yi < 0 || yi >= RT) continue;
        #pragma unroll
        for (int kx = 0; kx < 4; ++kx) {
            int xi = x + kx - 1;
            if (xi < 0 || xi >= RT) continue;
            sum += (k1[ky] * k1[kx] * (1.f / 16.f)) * Tb[yi * RT + xi];
        }
    }
    sum += nw_p[0] * noise[y * R2 + x] + ab[o];
    out[idx] = lrelu(sum);
}

// toRGB: 1x1 modulated conv without demod, + bias
__global__ __launch_bounds__(256) void k_torgb(
    const float* __restrict__ x, const float* __restrict__ s,
    const float* __restrict__ Wrgb, const float* __restrict__ bias,
    float* __restrict__ out, int IC, int NP)
{
    long idx = (long)blockIdx.x * blockDim.x + threadIdx.x;
    long total = (long)BATCH * 3 * NP;
    if (idx >= total) return;
    int p = idx % NP;
    int v = (idx / NP) % 3;
    int b = idx / ((long)3 * NP);
    float invs = rsqrtf((float)IC);
    const float* xb = x + (long)b * IC * NP;
    float acc = 0.f;
    for (int i = 0; i < IC; ++i)
        acc += Wrgb[v * IC + i] * s[b * IC + i] * xb[(long)i * NP + p];
    out[idx] = acc * invs + bias[v];
}

// skip += upfirdn(skip_old, K2D*4, up=2, pad=(2,1)):  Rs -> R2=2*Rs
__global__ __launch_bounds__(256) void k_skip_upfir_add(
    const float* __restrict__ old, float* __restrict__ out, int Rs, int R2)
{
    long idx = (long)blockIdx.x * blockDim.x + threadIdx.x;
    long total = (long)BATCH * 3 * R2 * R2;
    if (idx >= total) return;
    int x = idx % R2;
    int y = (idx / R2) % R2;
    int v = (idx / ((long)R2 * R2)) % 3;
    int b = idx / ((long)R2 * R2 * 3);
    const float k1[4] = {1.f, 3.f, 3.f, 1.f};
    const float* ob = old + ((long)b * 3 + v) * Rs * Rs;
    float sum = 0.f;
    #pragma unroll
    for (int ky = 0; ky < 4; ++ky) {
        int u = y + ky - 2;
        if (u < 0 || u > 2 * Rs - 2 || (u & 1)) continue;
        #pragma unroll
        for (int kx = 0; kx < 4; ++kx) {
            int w = x + kx - 2;
            if (w < 0 || w > 2 * Rs - 2 || (w & 1)) continue;
            sum += (k1[ky] * k1[kx] * (1.f / 16.f)) * ob[(u >> 1) * Rs + (w >> 1)];
        }
    }
    out[idx] += sum;
}

// ---------------------------------------------------------------------------
extern "C" void kernel_launch(void* const* d_in, const int* in_sizes, int n_in,
                              void* d_out, int out_size, void* d_ws, size_t ws_size,
                              hipStream_t stream)
{
    (void)in_sizes; (void)n_in; (void)out_size; (void)ws_size;

    const float* z      = (const float*)d_in[0];
    const float* cond   = (const float*)d_in[1];
    const float* lam    = (const float*)d_in[2];
    const float* noz[13];
    for (int j = 0; j < 13; ++j) noz[j] = (const float*)d_in[3 + j];
    const float* socket_w = (const float*)d_in[16];
    const float* socket_b = (const float*)d_in[17];
    const float* mlpw[9]; const float* mlpb[9];
    for (int l = 0; l < 9; ++l) { mlpw[l] = (const float*)d_in[18 + 2 * l];
                                  mlpb[l] = (const float*)d_in[19 + 2 * l]; }
    const float* cconst = (const float*)d_in[36];
    // conv1 {W,mw,mb,nw,ab} @37..41 ; torgb1 {W,mw,mb,bias} @42..45
    // convs[j] {W,mw,mb,nw,ab} @46+5j ; torgbs[t] {W,mw,mb,bias} @106+4t

    // workspace carve-up
    size_t off = 0;
    auto alloc = [&](size_t elems) -> float* {
        off = (off + 255) & ~(size_t)255;
        float* p = (float*)((char*)d_ws + off);
        off += elems * sizeof(float);
        return p;
    };
    float* latA  = alloc(BATCH * STYLE);
    float* latB  = alloc(BATCH * STYLE);
    float* ybuf  = alloc(BATCH * STYLE);
    float* sbuf  = alloc(BATCH * STYLE);
    float* dbuf  = alloc(BATCH * STYLE);
    float* srgb  = alloc(BATCH * STYLE);
    float* skipA = alloc((size_t)BATCH * 3 * 256 * 256);
    const size_t XELEMS = (size_t)BATCH * 128 * 257 * 257; // largest (T @256 level)
    float* X0 = alloc(XELEMS);
    float* X1 = alloc(XELEMS);
    float* X2 = alloc(XELEMS);

    const float inv_s512  = rsqrtf(512.f);
    const float mlp_scale = 0.01f * inv_s512;

    auto lin = [&](const float* in, const float* w, const float* bia, float* o,
                   int IN, int OUT, float sc, float lr, int act) {
        int th = BATCH * OUT;
        k_linear<<<(th + 255) / 256, 256, 0, stream>>>(in, w, bia, o, IN, OUT, sc, lr, act);
    };

    // ---- mapping network ----
    lin(cond, socket_w, socket_b, ybuf, 3 * 32 * 32, STYLE, rsqrtf(3072.f), 1.f, 0);
    k_pixelnorm<<<BATCH, 256, 0, stream>>>(z, latA);
    float* cur = latA; float* nxt = latB;
    for (int l = 0; l < 3; ++l) {
        lin(cur, mlpw[l], mlpb[l], nxt, STYLE, STYLE, mlp_scale, 0.01f, 1);
        float* t = cur; cur = nxt; nxt = t;
    }
    k_mix<<<(BATCH * STYLE + 255) / 256, 256, 0, stream>>>(cur, ybuf, lam, nxt);
    { float* t = cur; cur = nxt; nxt = t; }
    for (int l = 3; l < 9; ++l) {
        lin(cur, mlpw[l], mlpb[l], nxt, STYLE, STYLE, mlp_scale, 0.01f, 1);
        float* t = cur; cur = nxt; nxt = t;
    }
    const float* lat = cur;

    auto run_conv = [&](const float* xin, long bstride, float* xout,
                        int base, int IC, int OC, int R, const float* noise) {
        const float* W  = (const float*)d_in[base + 0];
        const float* mw = (const float*)d_in[base + 1];
        const float* mb = (const float*)d_in[base + 2];
        const float* nw = (const float*)d_in[base + 3];
        const float* ab = (const float*)d_in[base + 4];
        lin(lat, mw, mb, sbuf, STYLE, IC, inv_s512, 1.f, 0);
        k_demod<<<(BATCH * OC + 255) / 256, 256, 0, stream>>>(W, sbuf, dbuf, IC, OC);
        int NP = R * R;
        dim3 grid((NP + 127) / 128, OC / 64, BATCH);
        k_modconv<false><<<grid, 256, 0, stream>>>(xin, bstride, sbuf, W, dbuf,
                                                   noise, nw, ab, xout, IC, OC, R);
    };

    auto run_upconv = [&](const float* xin, long bstride, float* Tbuf, float* xout,
                          int base, int IC, int OC, int R, const float* noise) {
        const float* W  = (const float*)d_in[base + 0];
        const float* mw = (const float*)d_in[base + 1];
        const float* mb = (const float*)d_in[base + 2];
        const float* nw = (const float*)d_in[base + 3];
        const float* ab = (const float*)d_in[base + 4];
        lin(lat, mw, mb, sbuf, STYLE, IC, inv_s512, 1.f, 0);
        k_demod<<<(BATCH * OC + 255) / 256, 256, 0, stream>>>(W, sbuf, dbuf, IC, OC);
        int RT = 2 * R + 1, R2 = 2 * R;
        int NPT = RT * RT;
        dim3 grid((NPT + 127) / 128, OC / 64, BATCH);
        k_modconv<true><<<grid, 256, 0, stream>>>(xin, bstride, sbuf, W, dbuf,
                                                  nullptr, nullptr, nullptr, Tbuf, IC, OC, R);
        long tot = (long)BATCH * OC * R2 * R2;
        k_blur_act<<<(tot + 255) / 256, 256, 0, stream>>>(Tbuf, OC, RT, R2,
                                                          noise, nw, ab, xout);
    };

    auto run_torgb = [&](const float* x, float* rgb, int base, int IC, int R) {
        const float* W  = (const float*)d_in[base + 0];
        const float* mw = (const float*)d_in[base + 1];
        const float* mb = (const float*)d_in[base + 2];
        const float* bi = (const float*)d_in[base + 3];
        lin(lat, mw, mb, srgb, STYLE, IC, inv_s512, 1.f, 0);
        long tot = (long)BATCH * 3 * R * R;
        k_torgb<<<(tot + 255) / 256, 256, 0, stream>>>(x, srgb, W, bi, rgb, IC, R * R);
    };

    // ---- synthesis ----
    // conv1 @ 4x4 on broadcast const (batch stride 0)
    run_conv(cconst, 0, X0, 37, 512, 512, 4, noz[0]);
    run_torgb(X0, skipA, 42, 512, 4);

    const int icA[6] = {512, 512, 512, 512, 512, 256};
    const int ocA[6] = {512, 512, 512, 512, 256, 128};
    float* xin = X0; float* tbuf = X1; float* xmid = X2;
    int ni = 1;
    for (int bi = 0; bi < 6; ++bi) {
        int R  = 4 << bi;       // input res
        int R2 = 8 << bi;       // output res
        int IC = icA[bi], OC = ocA[bi];
        int cb = 46 + 10 * bi;                 // convs[2*bi]
        run_upconv(xin, (long)IC * R * R, tbuf, xmid, cb, IC, OC, R, noz[ni]);
        // second conv: reuse xin region as destination (old activation dead)
        run_conv(xmid, (long)OC * R2 * R2, xin, cb + 5, OC, OC, R2, noz[ni + 1]);
        float* xcur = xin;
        // torgb + skip upsample-add; final level writes straight to d_out
        float* rgb_dst = (bi == 5) ? (float*)d_out : tbuf; // tbuf free now, reuse as rgb
        run_torgb(xcur, rgb_dst, 106 + 4 * bi, OC, R2);
        long tot = (long)BATCH * 3 * R2 * R2;
        k_skip_upfir_add<<<(tot + 255) / 256, 256, 0, stream>>>(skipA, rgb_dst, R, R2);
        if (bi < 5) {
            // rgb_dst (== tbuf) now holds new skip
            float* t = skipA; skipA = rgb_dst; rgb_dst = t;
            float* nin = xcur;          // next input
            float* nt  = xmid;          // next T buffer (xmid dead after conv2)
            float* nm  = (nin == X0) ? ((nt == X1) ? X2 : X1)
                                     : ((nin == X1) ? ((nt == X0) ? X2 : X0)
                                                    : ((nt == X0) ? X1 : X0));
            xin = nin; tbuf = nt; xmid = nm;
        }
        ni += 2;
    }
}